// JacobianRegulariser2D_23441931502078
// MI455X (gfx1250) — compile-verified
//
#include <hip/hip_runtime.h>

// Problem geometry (from reference): displacement (1, 2, 4096, 4096) f32.
// Output pixels: rows/cols 2..4093 (4092 x 4092). Mean of relu(-jac_det)^2.
#define W      4096
#define NROWS  4092   // output rows, r = 2 .. 4093
#define NPART  4092   // one partial per row-block

typedef float v2f __attribute__((ext_vector_type(2)));
typedef float v8f __attribute__((ext_vector_type(8)));

// ---------------------------------------------------------------------------
// Pass 1: one block per output row. 256 threads x 4 float4-groups cover all
// 4096 columns with 16B-aligned b128 loads; crop mask zeroes cols <2 / >4093.
// Streams 3 rows x 2 channels; whole tensor fits in 192MB L2 so DRAM traffic
// is ~134MB total -> ~5.8us bandwidth floor on MI455X.
// ---------------------------------------------------------------------------
__global__ __launch_bounds__(256) void jacreg_rows(const float* __restrict__ disp,
                                                   float* __restrict__ partial) {
  const int tid = threadIdx.x;
  const int r = blockIdx.x + 2;                 // output row, 2..4093
  const float* ch0 = disp;
  const float* ch1 = disp + (size_t)W * W;
  const size_t rowm = (size_t)(r - 1) * W;
  const size_t rowc = (size_t)r * W;
  const size_t rowp = (size_t)(r + 1) * W;

  float acc = 0.0f;
#pragma unroll
  for (int k = 0; k < 4; ++k) {
    const int c4 = (tid + k * 256) * 4;         // 16B-aligned base column
    const float4 u0m = *(const float4*)(ch0 + rowm + c4);
    const float4 u0p = *(const float4*)(ch0 + rowp + c4);
    const float4 u0c = *(const float4*)(ch0 + rowc + c4);
    const float4 u1m = *(const float4*)(ch1 + rowm + c4);
    const float4 u1p = *(const float4*)(ch1 + rowp + c4);
    const float4 u1c = *(const float4*)(ch1 + rowc + c4);
    // x-neighbors just outside the vector; clamp is safe because the pixels
    // that would need true col -1 / 4096 are outside the crop and masked.
    const int cl = (c4 > 0) ? c4 - 1 : 0;
    const int cr = (c4 + 4 < W) ? c4 + 4 : W - 1;
    const float u0l = ch0[rowc + cl], u0r = ch0[rowc + cr];
    const float u1l = ch1[rowc + cl], u1r = ch1[rowc + cr];

    const float c0[6] = {u0l, u0c.x, u0c.y, u0c.z, u0c.w, u0r};
    const float c1[6] = {u1l, u1c.x, u1c.y, u1c.z, u1c.w, u1r};
    const float m0[4] = {u0m.x, u0m.y, u0m.z, u0m.w};
    const float p0[4] = {u0p.x, u0p.y, u0p.z, u0p.w};
    const float m1[4] = {u1m.x, u1m.y, u1m.z, u1m.w};
    const float p1[4] = {u1p.x, u1p.y, u1p.z, u1p.w};

#pragma unroll
    for (int i = 0; i < 4; ++i) {
      const int col = c4 + i;
      // Conv2d [0.5,0,-0.5] cross-correlation: (x[i-1]-x[i+1])*0.5
      const float dy0 = (m0[i] - p0[i]) * 0.5f;   // d/dy channel 0
      const float dy1 = (m1[i] - p1[i]) * 0.5f;   // d/dy channel 1
      const float dx0 = (c0[i] - c0[i + 2]) * 0.5f; // d/dx channel 0
      const float dx1 = (c1[i] - c1[i + 2]) * 0.5f; // d/dx channel 1
      // det = (dy0+1)*(dx1+1) - dx0*dy1
      const float det = (dy0 + 1.0f) * (dx1 + 1.0f) - dx0 * dy1;
      const float neg = fmaxf(-det, 0.0f);
      float v = neg * neg;
      if (col < 2 || col > W - 3) v = 0.0f;       // crop mask
      acc += v;
    }
  }

  __shared__ float sm[256];
  sm[tid] = acc;
  __syncthreads();
#pragma unroll
  for (int s = 128; s > 0; s >>= 1) {
    if (tid < s) sm[tid] += sm[tid + s];
    __syncthreads();
  }
  if (tid == 0) partial[blockIdx.x] = sm[0];
}

// ---------------------------------------------------------------------------
// Pass 2: single block folds the 4092 row partials. The last 256->1 stage is
// done on the matrix pipe: V_WMMA_F32_16X16X4_F32 with B = ones sums 64 f32
// per issue (D[m][n] = sum_k A[m][k] + C[m][n]); 4 chained issues consume all
// 256 values, then an 8-VGPR lane sum + shfl_xor(16) folds the 16x16 C tile.
// All-f32 exact math, deterministic, and exercises the gfx1250 WMMA path.
// ---------------------------------------------------------------------------
__global__ __launch_bounds__(256) void jacreg_final(const float* __restrict__ partial,
                                                    int n, float* __restrict__ out,
                                                    float inv_count) {
  __shared__ float sm[256];
  const int tid = threadIdx.x;
  float s = 0.0f;
  for (int i = tid; i < n; i += 256) s += partial[i];
  sm[tid] = s;
  __syncthreads();

  if (tid < 32) {                                 // wave 0 only: EXEC all-1s
    v2f b; b.x = 1.0f; b.y = 1.0f;                // B(4x16) = ones
    v8f c = {};                                   // C(16x16) accumulator
#pragma unroll
    for (int i = 0; i < 4; ++i) {
      v2f a;
      a.x = sm[i * 64 + tid];                     // any bijection into A(16x4)
      a.y = sm[i * 64 + 32 + tid];                // works: B=ones sums all 64
      c = __builtin_amdgcn_wmma_f32_16x16x4_f32(
          /*neg_a=*/false, a, /*neg_b=*/false, b,
          /*c_mod=*/(short)0, c, /*reuse_a=*/false, /*reuse_b=*/false);
    }
    // lane l holds D[0..7][l] (l<16) or D[8..15][l-16]: fold VGPRs, then pair.
    float t = c[0] + c[1] + c[2] + c[3] + c[4] + c[5] + c[6] + c[7];
    t += __shfl_xor(t, 16, 32);                   // rows 0-7 + rows 8-15
    if (tid == 0) out[0] = t * inv_count;
  }
}

extern "C" void kernel_launch(void* const* d_in, const int* in_sizes, int n_in,
                              void* d_out, int out_size, void* d_ws, size_t ws_size,
                              hipStream_t stream) {
  const float* disp = (const float*)d_in[0];
  float* out = (float*)d_out;
  float* part = (float*)d_ws;                     // NPART * 4 bytes scratch

  jacreg_rows<<<NROWS, 256, 0, stream>>>(disp, part);
  // 4092*4092 = 16744464 < 2^24, exactly representable in f32.
  jacreg_final<<<1, 256, 0, stream>>>(part, NPART, out,
                                      1.0f / (4092.0f * 4092.0f));
}